// PICOContrastiveRAG_37538014167097
// MI455X (gfx1250) — compile-verified
//
#include <hip/hip_runtime.h>
#include <math.h>

typedef __attribute__((ext_vector_type(2))) float v2f;
typedef __attribute__((ext_vector_type(8))) float v8f;

#define TOPK 5
#define E_DIM 768

// Issue one 16-byte async global->LDS copy (CDNA5 GLOBAL_LOAD_ASYNC_TO_LDS_B128).
// ldsPtr/gPtr must be 16-byte aligned. Completion tracked by ASYNCcnt.
__device__ __forceinline__ void async_copy16(void* ldsPtr, const void* gPtr)
{
    unsigned lo = (unsigned)(size_t)ldsPtr;               // LDS byte offset (addr[31:0])
    unsigned long long ga = (unsigned long long)(size_t)gPtr;
    asm volatile("global_load_async_to_lds_b128 %0, %1, off"
                 :: "v"(lo), "v"(ga) : "memory");
}
__device__ __forceinline__ void async_wait0()
{
    asm volatile("s_wait_asynccnt 0x0" ::: "memory");
}

// ---------------------------------------------------------------------------
// Generic small linear kernel: out = act( bias + in0@W0 + in1@W1 + in2@W2 )
// Weights are row-major [K, O]; slices of a larger weight keep stride O.
// ---------------------------------------------------------------------------
__global__ void lin3_kernel(const float* __restrict__ in0, const float* __restrict__ W0, int K0,
                            const float* __restrict__ in1, const float* __restrict__ W1, int K1,
                            const float* __restrict__ in2, const float* __restrict__ W2, int K2,
                            const float* __restrict__ bias, float* __restrict__ out,
                            int B, int O, int act)
{
    int t = blockIdx.x * blockDim.x + threadIdx.x;
    if (t >= B * O) return;
    int b = t / O, o = t % O;
    float s = bias[o];
    if (in0) { const float* ir = in0 + (size_t)b * K0;
               for (int k = 0; k < K0; ++k) s += ir[k] * W0[(size_t)k * O + o]; }
    if (in1) { const float* ir = in1 + (size_t)b * K1;
               for (int k = 0; k < K1; ++k) s += ir[k] * W1[(size_t)k * O + o]; }
    if (in2) { const float* ir = in2 + (size_t)b * K2;
               for (int k = 0; k < K2; ++k) s += ir[k] * W2[(size_t)k * O + o]; }
    if (act) s = fmaxf(s, 0.f);
    out[(size_t)b * O + o] = s;
}

// ---------------------------------------------------------------------------
// out[row] = l2norm(a[row] - b[row]) ; one block per row
// ---------------------------------------------------------------------------
__global__ void diff_l2norm_kernel(const float* __restrict__ a, const float* __restrict__ bm,
                                   float* __restrict__ out, int D)
{
    __shared__ float red[256];
    int row = blockIdx.x;
    float s = 0.f;
    for (int k = threadIdx.x; k < D; k += blockDim.x) {
        float d = a[(size_t)row * D + k] - bm[(size_t)row * D + k];
        s += d * d;
    }
    red[threadIdx.x] = s;
    __syncthreads();
    for (int off = blockDim.x >> 1; off > 0; off >>= 1) {
        if ((int)threadIdx.x < off) red[threadIdx.x] += red[threadIdx.x + off];
        __syncthreads();
    }
    float inv = 1.f / fmaxf(sqrtf(red[0]), 1e-12f);
    for (int k = threadIdx.x; k < D; k += blockDim.x)
        out[(size_t)row * D + k] = (a[(size_t)row * D + k] - bm[(size_t)row * D + k]) * inv;
}

// ---------------------------------------------------------------------------
// Fused corpus pass, one HBM pass over the corpus.
// Block = 128 threads (4 waves), 64 docs/tile, 16 docs (M=16) per wave.
// nw1 and q are staged through LDS with async global->LDS b128 copies and
// shared by all 4 waves; WMMA = v_wmma_f32_16x16x4_f32 (full f32 precision,
// required for stable top-k indices).
// ---------------------------------------------------------------------------
#define SMEM_FLOATS 12352   // max(nw1 chunk 64*128=8192, q stage 16*772=12352, ldsW 4096)
#define QSTRIDE 772         // 768 + 4 pad: lanes hit distinct LDS bank pairs

__global__ __launch_bounds__(128)
void corpus_score_kernel(const float* __restrict__ corpus,   // [nDocs, 768]
                         const float* __restrict__ q,        // [64, 768] unit rows
                         const float* __restrict__ nw1,      // [768, 128]
                         const float* __restrict__ nb1,      // [128]
                         const float* __restrict__ nw2,      // [128]
                         const float* __restrict__ nb2,      // [1]
                         float* __restrict__ candS,          // [64, nTiles, 5]
                         int*   __restrict__ candI,          // [64, nTiles, 5]
                         int nDocs, int nTiles)
{
    __shared__ float ldsInv[64];
    __shared__ float smem[SMEM_FLOATS];   // time-shared: B-stage / Q-stage / weighted

    const int tid  = threadIdx.x;
    const int wave = tid >> 5;
    const int lane = tid & 31;
    const int l15  = lane & 15;
    const int half = lane >> 4;           // selects K pair within 16x16x4 layouts
    const int tileBase = blockIdx.x * 64;

    int rowA = tileBase + wave * 16 + l15;
    if (rowA >= nDocs) rowA = nDocs - 1;  // clamp (OOB docs skipped in scan)
    const float* __restrict__ Arow = corpus + (size_t)rowA * E_DIM;

    // ---- per-doc inverse L2 norm (row split across lane halves) ----
    {
        float s = 0.f;
        const float* rp = Arow + half * (E_DIM / 2);
        for (int k = 0; k < E_DIM / 2; ++k) s += rp[k] * rp[k];
        s += __shfl_xor(s, 16, 32);
        if (lane < 16) ldsInv[wave * 16 + lane] = 1.f / fmaxf(sqrtf(s), 1e-12f);
    }

    // ======================= nuance GEMM =======================
    // H[16,128] = corpus_tile[16,768] @ nw1[768,128], K chunked by 64 through LDS.
    v8f acc[8] = {};
    const float* __restrict__ Abase = Arow + 2 * half;
#pragma unroll 1
    for (int kc = 0; kc < E_DIM / 64; ++kc) {
        // async-stage nw1[kc*64 .. +64][0..128]  (32 KB, contiguous) into smem
        {
            const float* src = nw1 + (size_t)kc * 64 * 128;
            for (int i = tid; i < 2048; i += 128)
                async_copy16(smem + (size_t)i * 4, src + (size_t)i * 4);
            async_wait0();
        }
        __syncthreads();
        const int k0 = kc * 64;
#pragma unroll 1
        for (int kk = 0; kk < 64; kk += 4) {
            v2f a = *reinterpret_cast<const v2f*>(Abase + k0 + kk);
            const float* Bk0 = smem + (size_t)(kk + 2 * half) * 128 + l15;
            const float* Bk1 = Bk0 + 128;
#pragma unroll
            for (int nb = 0; nb < 8; ++nb) {
                v2f bv; bv[0] = Bk0[nb * 16]; bv[1] = Bk1[nb * 16];
                acc[nb] = __builtin_amdgcn_wmma_f32_16x16x4_f32(
                    false, a, false, bv, (short)0, acc[nb], false, false);
            }
        }
        __syncthreads();   // before next chunk overwrites smem
    }

    // ---- bias + relu + @nw2 + sigmoid, reduced per doc (C/D row layout) ----
    float nb1v[8], nw2v[8];
#pragma unroll
    for (int nb = 0; nb < 8; ++nb) {
        nb1v[nb] = nb1[nb * 16 + l15];
        nw2v[nb] = nw2[nb * 16 + l15];
    }
    const float nb2s = nb2[0];

    float nu[8];   // per accumulator row r: doc = wave*16 + r + 8*half
#pragma unroll
    for (int r = 0; r < 8; ++r) {
        float s = 0.f;
#pragma unroll
        for (int nb = 0; nb < 8; ++nb)
            s += fmaxf(acc[nb][r] + nb1v[nb], 0.f) * nw2v[nb];
        s += __shfl_xor(s, 1, 32);
        s += __shfl_xor(s, 2, 32);
        s += __shfl_xor(s, 4, 32);
        s += __shfl_xor(s, 8, 32);
        nu[r] = 1.f / (1.f + expf(-(s + nb2s)));
    }

    // ======================= sim GEMM =======================
    // S[16,64] = corpus_tile[16,768] @ q^T, 16 queries at a time through LDS.
    v8f sacc[4] = {};
#pragma unroll 1
    for (int qc = 0; qc < 4; ++qc) {
        // async-stage q rows [qc*16, +16) into smem, row stride QSTRIDE floats
        {
            const float* src = q + (size_t)qc * 16 * E_DIM;
            for (int i = tid; i < 16 * 192; i += 128) {
                int row = i / 192, seg = i % 192;   // seg = 4-float segment
                async_copy16(smem + (size_t)row * QSTRIDE + seg * 4,
                             src  + (size_t)row * E_DIM   + seg * 4);
            }
            async_wait0();
        }
        __syncthreads();
#pragma unroll 1
        for (int kk = 0; kk < E_DIM; kk += 4) {
            v2f a  = *reinterpret_cast<const v2f*>(Abase + kk);
            v2f bv = *reinterpret_cast<const v2f*>(smem + (size_t)l15 * QSTRIDE + kk + 2 * half);
            sacc[qc] = __builtin_amdgcn_wmma_f32_16x16x4_f32(
                false, a, false, bv, (short)0, sacc[qc], false, false);
        }
        __syncthreads();
    }

    // ---- weighted = sim * invnorm * nuance -> smem as [doc_local][query] ----
    float invn[8];
#pragma unroll
    for (int r = 0; r < 8; ++r) invn[r] = ldsInv[wave * 16 + r + 8 * half];
#pragma unroll
    for (int qb = 0; qb < 4; ++qb) {
#pragma unroll
        for (int r = 0; r < 8; ++r) {
            int dl = wave * 16 + r + 8 * half;
            smem[dl * 64 + qb * 16 + l15] = sacc[qb][r] * invn[r] * nu[r];
        }
    }
    __syncthreads();

    // ---- per-query local top-5 over this tile (docs in increasing order) ----
    if (tid < 64) {
        const int qx = tid;
        float ts[TOPK]; int ti[TOPK];
#pragma unroll
        for (int j = 0; j < TOPK; ++j) { ts[j] = -INFINITY; ti[j] = 0; }
        for (int d = 0; d < 64; ++d) {
            if (tileBase + d >= nDocs) break;
            float v = smem[d * 64 + qx];
            if (v > ts[TOPK - 1]) {
                int p = TOPK - 1;
                while (p > 0 && v > ts[p - 1]) { ts[p] = ts[p - 1]; ti[p] = ti[p - 1]; --p; }
                ts[p] = v; ti[p] = tileBase + d;
            }
        }
        size_t base = ((size_t)qx * nTiles + blockIdx.x) * TOPK;
#pragma unroll
        for (int j = 0; j < TOPK; ++j) { candS[base + j] = ts[j]; candI[base + j] = ti[j]; }
    }
}

// ---------------------------------------------------------------------------
// Final deterministic top-5 merge per query + gather retrieved embeddings.
// One block per query; tiles scanned in increasing order -> jax tie behavior.
// ---------------------------------------------------------------------------
__global__ void topk_final_kernel(const float* __restrict__ candS, const int* __restrict__ candI,
                                  int nTiles, const float* __restrict__ corpus,
                                  float* __restrict__ outScores, int* __restrict__ outIdx,
                                  float* __restrict__ flatR)
{
    __shared__ int sIdx[TOPK];
    const int b = blockIdx.x;
    if (threadIdx.x == 0) {
        float ts[TOPK]; int ti[TOPK];
#pragma unroll
        for (int j = 0; j < TOPK; ++j) { ts[j] = -INFINITY; ti[j] = 0; }
        const float* cs = candS + (size_t)b * nTiles * TOPK;
        const int*   ci = candI + (size_t)b * nTiles * TOPK;
        const int n = nTiles * TOPK;
        for (int t = 0; t < n; ++t) {
            float v = cs[t];
            if (v > ts[TOPK - 1]) {
                int p = TOPK - 1;
                while (p > 0 && v > ts[p - 1]) { ts[p] = ts[p - 1]; ti[p] = ti[p - 1]; --p; }
                ts[p] = v; ti[p] = ci[t];
            }
        }
#pragma unroll
        for (int j = 0; j < TOPK; ++j) {
            outScores[b * TOPK + j] = ts[j];
            outIdx[b * TOPK + j]    = ti[j];
            sIdx[j] = ti[j];
        }
    }
    __syncthreads();
    for (int j = 0; j < TOPK; ++j) {
        const float* src = corpus + (size_t)sIdx[j] * E_DIM;
        for (int e = threadIdx.x; e < E_DIM; e += blockDim.x)
            flatR[(size_t)b * (E_DIM * TOPK) + (size_t)j * E_DIM + e] = src[e];
    }
}

// ---------------------------------------------------------------------------
extern "C" void kernel_launch(void* const* d_in, const int* in_sizes, int n_in,
                              void* d_out, int out_size, void* d_ws, size_t ws_size,
                              hipStream_t stream)
{
    (void)n_in; (void)out_size; (void)ws_size;
    const float* patient   = (const float*)d_in[0];
    const float* treatment = (const float*)d_in[1];
    const float* confound  = (const float*)d_in[2];
    /* d_in[3] = outcome_features : unused downstream */
    const float* corpus    = (const float*)d_in[4];
    const float* popW1=(const float*)d_in[5],  *popB1=(const float*)d_in[6],
               * popW2=(const float*)d_in[7],  *popB2=(const float*)d_in[8];
    const float* intW1=(const float*)d_in[9],  *intB1=(const float*)d_in[10],
               * intW2=(const float*)d_in[11], *intB2=(const float*)d_in[12];
    const float* cmpB1=(const float*)d_in[14],
               * cmpW2=(const float*)d_in[15], *cmpB2=(const float*)d_in[16];
    /* d_in[13] cmpW1 unused (input is zeros); d_in[17..20] outp unused */
    const float* effW1=(const float*)d_in[21], *effB1=(const float*)d_in[22],
               * effW2=(const float*)d_in[23], *effB2=(const float*)d_in[24];
    const float* nuW1 =(const float*)d_in[25], *nuB1 =(const float*)d_in[26],
               * nuW2 =(const float*)d_in[27], *nuB2 =(const float*)d_in[28];
    const float* trW=(const float*)d_in[29], *trB=(const float*)d_in[30];
    const float* cfW=(const float*)d_in[31], *cfB=(const float*)d_in[32];
    const float* reW=(const float*)d_in[33], *reB=(const float*)d_in[34];
    const float* op1W=(const float*)d_in[35], *op1B=(const float*)d_in[36];
    const float* op2W=(const float*)d_in[37], *op2B=(const float*)d_in[38];
    const float* op3W=(const float*)d_in[39], *op3B=(const float*)d_in[40];

    const int B = 64, H = 256, E = E_DIM, OUTD = 16;
    const int nDocs  = in_sizes[4] / E;
    const int nTiles = (nDocs + 63) / 64;

    // ---- workspace carve-up ----
    float* w = (float*)d_ws;
    size_t off = 0;
    float* hid   = w + off; off += (size_t)B * H;
    float* pop   = w + off; off += (size_t)B * H;
    float* inter = w + off; off += (size_t)B * H;
    float* cmpv  = w + off; off += (size_t)B * H;
    float* effpi = w + off; off += (size_t)B * E;
    float* effpc = w + off; off += (size_t)B * E;
    float* trO   = w + off; off += (size_t)B * H;
    float* cfO   = w + off; off += (size_t)B * H;
    float* reO   = w + off; off += (size_t)B * H;
    float* h1    = w + off; off += (size_t)B * H;
    float* h2    = w + off; off += (size_t)B * (H / 2);
    float* flatR = w + off; off += (size_t)B * E * TOPK;
    float* candS = w + off; off += (size_t)B * nTiles * TOPK;
    int*   candI = (int*)(w + off);

    // ---- output layout: outcome[64,16] | scores[64,5] | idx[64,5] | cemb[64,768] ----
    float* outF    = (float*)d_out;
    float* outcome = outF;
    float* scores  = outF + B * OUTD;
    int*   idxOut  = (int*)outF + B * OUTD + B * TOPK;
    float* cemb    = outF + B * OUTD + 2 * B * TOPK;   // also used as query matrix q

    dim3 blk(256);
    auto g = [](int n) { return dim3((unsigned)((n + 255) / 256)); };

    // PICO MLPs (treat_p zero-padded -> K=128; control_p == 0 -> hidden = relu(b1))
    lin3_kernel<<<g(B*H), blk, 0, stream>>>(patient, popW1, 512, nullptr,nullptr,0, nullptr,nullptr,0, popB1, hid,  B, H, 1);
    lin3_kernel<<<g(B*H), blk, 0, stream>>>(hid,     popW2, H,   nullptr,nullptr,0, nullptr,nullptr,0, popB2, pop,  B, H, 0);
    lin3_kernel<<<g(B*H), blk, 0, stream>>>(treatment, intW1, 128, nullptr,nullptr,0, nullptr,nullptr,0, intB1, hid,  B, H, 1);
    lin3_kernel<<<g(B*H), blk, 0, stream>>>(hid,     intW2, H,   nullptr,nullptr,0, nullptr,nullptr,0, intB2, inter, B, H, 0);
    lin3_kernel<<<g(B*H), blk, 0, stream>>>(nullptr, nullptr, 0, nullptr,nullptr,0, nullptr,nullptr,0, cmpB1, hid,  B, H, 1);
    lin3_kernel<<<g(B*H), blk, 0, stream>>>(hid,     cmpW2, H,   nullptr,nullptr,0, nullptr,nullptr,0, cmpB2, cmpv, B, H, 0);

    // effect encoder on [pop|inter] and [pop|cmpv]
    lin3_kernel<<<g(B*H), blk, 0, stream>>>(pop, effW1, H, inter, effW1 + (size_t)H * H, H, nullptr,nullptr,0, effB1, hid, B, H, 1);
    lin3_kernel<<<g(B*E), blk, 0, stream>>>(hid, effW2, H, nullptr,nullptr,0, nullptr,nullptr,0, effB2, effpi, B, E, 0);
    lin3_kernel<<<g(B*H), blk, 0, stream>>>(pop, effW1, H, cmpv,  effW1 + (size_t)H * H, H, nullptr,nullptr,0, effB1, hid, B, H, 1);
    lin3_kernel<<<g(B*E), blk, 0, stream>>>(hid, effW2, H, nullptr,nullptr,0, nullptr,nullptr,0, effB2, effpc, B, E, 0);

    // cemb = l2norm(effpi - effpc)  (q == l2norm(cemb) == cemb)
    diff_l2norm_kernel<<<dim3(B), blk, 0, stream>>>(effpi, effpc, cemb, E);

    // fused corpus scoring pass (WMMA + async LDS staging) + deterministic top-K
    corpus_score_kernel<<<dim3(nTiles), dim3(128), 0, stream>>>(
        corpus, cemb, nuW1, nuB1, nuW2, nuB2, candS, candI, nDocs, nTiles);
    topk_final_kernel<<<dim3(B), blk, 0, stream>>>(candS, candI, nTiles, corpus, scores, idxOut, flatR);

    // retrieval head
    lin3_kernel<<<g(B*H), blk, 0, stream>>>(treatment, trW, 128, nullptr,nullptr,0, nullptr,nullptr,0, trB, trO, B, H, 0);
    lin3_kernel<<<g(B*H), blk, 0, stream>>>(confound,  cfW, 384, nullptr,nullptr,0, nullptr,nullptr,0, cfB, cfO, B, H, 0);
    lin3_kernel<<<g(B*H), blk, 0, stream>>>(flatR,     reW, E * TOPK, nullptr,nullptr,0, nullptr,nullptr,0, reB, reO, B, H, 0);
    lin3_kernel<<<g(B*H), blk, 0, stream>>>(trO, op1W, H, cfO, op1W + (size_t)H * H, H, reO, op1W + (size_t)2 * H * H, H, op1B, h1, B, H, 1);
    lin3_kernel<<<g(B*(H/2)), blk, 0, stream>>>(h1, op2W, H,     nullptr,nullptr,0, nullptr,nullptr,0, op2B, h2, B, H / 2, 1);
    lin3_kernel<<<g(B*OUTD),  blk, 0, stream>>>(h2, op3W, H / 2, nullptr,nullptr,0, nullptr,nullptr,0, op3B, outcome, B, OUTD, 0);
}